// PlainGuidedUpsampler_20607253086409
// MI455X (gfx1250) — compile-verified
//
#include <hip/hip_runtime.h>
#include <math.h>

#define GH 224
#define GW 224
#define SH 14
#define SW 14
#define SC 64
#define DIL 16
#define KWIN 7
#define KTOT 49          // 7x7 coarse window
#define KPAD 52          // padded to multiple of 4 (WMMA K-step = 4)
#define NPIX 128         // 16 wide x 8 tall pixel tile per workgroup

typedef __attribute__((ext_vector_type(2))) float v2f;
typedef __attribute__((ext_vector_type(8))) float v8f;

__device__ __forceinline__ int reflect224(int t) {
    t = (t < 0) ? -t : t;
    t = (t > 223) ? (446 - t) : t;
    return t;
}

// Per-batch std of guidance, ddof=1 (torch.std unbiased). One block per batch.
__global__ __launch_bounds__(256)
void sigma_kernel(const float* __restrict__ g, float* __restrict__ sig) {
    const int b = blockIdx.x;
    const float* p = g + (size_t)b * 3 * GH * GW;
    const int N = 3 * GH * GW;
    float s = 0.f, s2 = 0.f;
    for (int i = threadIdx.x; i < N; i += 256) {
        float v = p[i];
        s += v;
        s2 += v * v;
    }
    __shared__ float sh[256], sh2[256];
    sh[threadIdx.x] = s;
    sh2[threadIdx.x] = s2;
    __syncthreads();
    for (int o = 128; o > 0; o >>= 1) {
        if (threadIdx.x < o) {
            sh[threadIdx.x]  += sh[threadIdx.x + o];
            sh2[threadIdx.x] += sh2[threadIdx.x + o];
        }
        __syncthreads();
    }
    if (threadIdx.x == 0) {
        float mean = sh[0] / (float)N;
        float var  = (sh2[0] - sh[0] * mean) / (float)(N - 1);
        sig[b] = sqrtf(fmaxf(var, 0.f));
    }
}

// One workgroup per (batch, 8-row band, 16-col tile). 128 threads = 4 waves.
__global__ __launch_bounds__(128)
void bilateral_kernel(const float* __restrict__ src, const float* __restrict__ gd,
                      const float* __restrict__ sig, float* __restrict__ out) {
    // W stored as [kpair][pix][2] -> B operand = one aligned float2 LDS read
    __shared__ float Wl[(KPAD / 2) * NPIX * 2];   // 26*256 floats = 26.0 KB
    __shared__ float Sw[SC * KPAD];               // 64*52  floats = 13.0 KB

    const int t    = threadIdx.x;
    const int lane = t & 31;
    const int wid  = t >> 5;

    const int TX = blockIdx.x;        // 0..13  (16-col tile)
    const int by = blockIdx.y;        // 0..27  (8-row band)
    const int b  = blockIdx.z;        // batch
    const int TY = by >> 1;           // coarse 16x16 block row

    // shared coarse window base (covers all bilinear taps incl. reflection)
    const int ubase = min(max(TY - 3, 0), SH - KWIN);
    const int vbase = min(max(TX - 3, 0), SW - KWIN);

    // ---- stage source window: Sw[c][k], k = du*7+dv over [ubase..+6]x[vbase..+6]
    const float* sb = src + (size_t)b * SC * SH * SW;
    for (int idx = t; idx < SC * KPAD; idx += NPIX) {
        int c = idx / KPAD;
        int k = idx - c * KPAD;
        float v = 0.f;
        if (k < KTOT) {
            int du = k / KWIN, dv = k - du * KWIN;
            v = sb[c * (SH * SW) + (ubase + du) * SW + (vbase + dv)];
        }
        Sw[idx] = v;
    }

    // ---- zero my private weight column
    #pragma unroll
    for (int kk = 0; kk < KPAD / 2; ++kk) {
        Wl[kk * (NPIX * 2) + t * 2 + 0] = 0.f;
        Wl[kk * (NPIX * 2) + t * 2 + 1] = 0.f;
    }

    // ---- per-pixel bilateral weights (thread t <-> pixel (py,px))
    const int py = t >> 4;                 // 0..7
    const int px = t & 15;                 // 0..15
    const int y  = by * 8 + py;
    const int x  = TX * 16 + px;

    const float* gb = gd + (size_t)b * 3 * GH * GW;
    float g0 = gb[0 * GH * GW + y * GW + x];
    float g1 = gb[1 * GH * GW + y * GW + x];
    float g2 = gb[2 * GH * GW + y * GW + x];
    float ci = 1.f / fmaxf(sqrtf(g0 * g0 + g1 * g1 + g2 * g2), 1e-12f);
    g0 *= ci; g1 *= ci; g2 *= ci;

    const float sr = sig[b];
    const float inv2s2 = 1.f / (2.f * sr * sr);

    // spatial kernel exp(-(di^2+dj^2)/(2*2.5^2)), row-major (i,j)
    const float SP[25] = {
        0.52729242f, 0.67032005f, 0.72614904f, 0.67032005f, 0.52729242f,
        0.67032005f, 0.85214379f, 0.92311635f, 0.85214379f, 0.67032005f,
        0.72614904f, 0.92311635f, 1.00000000f, 0.92311635f, 0.72614904f,
        0.67032005f, 0.85214379f, 0.92311635f, 0.85214379f, 0.67032005f,
        0.52729242f, 0.67032005f, 0.72614904f, 0.67032005f, 0.52729242f};

    float w[25];
    float wsum = 0.f;
    #pragma unroll
    for (int p = 0; p < 25; ++p) {
        int i = p / 5, j = p % 5;
        int ry = reflect224(y + (i - 2) * DIL);
        int cx = reflect224(x + (j - 2) * DIL);
        float q0 = gb[0 * GH * GW + ry * GW + cx];
        float q1 = gb[1 * GH * GW + ry * GW + cx];
        float q2 = gb[2 * GH * GW + ry * GW + cx];
        float qi = 1.f / fmaxf(sqrtf(q0 * q0 + q1 * q1 + q2 * q2), 1e-12f);
        float rk = 2.f - (q0 * g0 + q1 * g1 + q2 * g2) * qi;
        float ww = __expf(-rk * rk * inv2s2) * SP[p];
        w[p] = ww;
        wsum += ww;
    }
    const float wni = 1.f / fmaxf(wsum, 1e-8f);

    // ---- scatter (25 patches x 4 bilinear taps) into my private column of W
    #pragma unroll
    for (int p = 0; p < 25; ++p) {
        int i = p / 5, j = p % 5;
        int ry = reflect224(y + (i - 2) * DIL);
        int cx = reflect224(x + (j - 2) * DIL);
        // bilinear (align_corners=False): f = (q + 0.5)/16 - 0.5
        float fy = ((float)ry - 7.5f) * 0.0625f;
        float fx = ((float)cx - 7.5f) * 0.0625f;
        int y0 = (int)floorf(fy);
        int x0 = (int)floorf(fx);
        float ty = fy - (float)y0, tx = fx - (float)x0;
        int u0 = max(y0, 0), u1 = min(y0 + 1, SH - 1);
        int v0 = max(x0, 0), v1 = min(x0 + 1, SW - 1);
        float wp = w[p] * wni;
        int lu0 = u0 - ubase, lu1 = u1 - ubase;
        int lv0 = v0 - vbase, lv1 = v1 - vbase;
        int k00 = lu0 * KWIN + lv0, k01 = lu0 * KWIN + lv1;
        int k10 = lu1 * KWIN + lv0, k11 = lu1 * KWIN + lv1;
        Wl[(k00 >> 1) * (NPIX * 2) + t * 2 + (k00 & 1)] += wp * (1.f - ty) * (1.f - tx);
        Wl[(k01 >> 1) * (NPIX * 2) + t * 2 + (k01 & 1)] += wp * (1.f - ty) * tx;
        Wl[(k10 >> 1) * (NPIX * 2) + t * 2 + (k10 & 1)] += wp * ty * (1.f - tx);
        Wl[(k11 >> 1) * (NPIX * 2) + t * 2 + (k11 & 1)] += wp * ty * tx;
    }

    __syncthreads();

    // ---- GEMM: out[c,pix] = sum_k Sw[c,k] * W[k,pix] via V_WMMA_F32_16X16X4_F32
    // 32 tiles = 4 channel-chunks x 8 pixel-rows; 4 waves -> 8 tiles each.
    const int m  = lane & 15;     // M index (channel within chunk / N within row)
    const int hf = lane >> 4;     // lane half selects K pair (A) / K pair (B), M+8 (D)
    for (int tIdx = wid; tIdx < 32; tIdx += 4) {
        int cchunk = tIdx >> 3;        // 0..3
        int prow   = tIdx & 7;         // 0..7 (pixel row within band)
        int c0 = cchunk * 16;
        v8f acc = {};
        #pragma unroll
        for (int ks = 0; ks < KPAD / 4; ++ks) {       // 13 K-steps of 4
            int kbase = ks * 4;
            // A (16x4): lane half 0 -> K = kbase+{0,1}; half 1 -> K = kbase+{2,3}
            v2f a = *(const v2f*)&Sw[(c0 + m) * KPAD + kbase + hf * 2];
            // B (4x16): VGPR0 rows {kbase, kbase+2}, VGPR1 rows {kbase+1, kbase+3}
            int kk = ks * 2 + hf;                     // kpair index
            v2f bv = *(const v2f*)&Wl[kk * (NPIX * 2) + (prow * 16 + m) * 2];
            acc = __builtin_amdgcn_wmma_f32_16x16x4_f32(
                false, a, false, bv, (short)0, acc, false, false);
        }
        // D (16x16 f32): VGPR r -> M = r (lanes 0-15) / r+8 (lanes 16-31); N = lane&15
        int oy = by * 8 + prow;
        int ox = TX * 16 + m;
        #pragma unroll
        for (int r = 0; r < 8; ++r) {
            int c = c0 + r + hf * 8;
            out[(((size_t)b * SC + c) * GH + oy) * GW + ox] = acc[r];
        }
    }
}

extern "C" void kernel_launch(void* const* d_in, const int* in_sizes, int n_in,
                              void* d_out, int out_size, void* d_ws, size_t ws_size,
                              hipStream_t stream) {
    const float* src = (const float*)d_in[0];   // (2,64,14,14)
    const float* gd  = (const float*)d_in[1];   // (2,3,224,224)
    float* out = (float*)d_out;                 // (2,64,224,224)
    float* sig = (float*)d_ws;                  // 2 floats

    sigma_kernel<<<dim3(2), dim3(256), 0, stream>>>(gd, sig);
    bilateral_kernel<<<dim3(GW / 16, GH / 8, 2), dim3(NPIX), 0, stream>>>(src, gd, sig, out);
}